// GAT_56255481643659
// MI455X (gfx1250) — compile-verified
//
#include <hip/hip_runtime.h>
#include <math.h>

#define N_NODES 40000
#define N_EDGES 640000
#define IN_CH   128
#define HID     32
#define HEADS   4
#define OUT_CH  112
#define NEG_SLOPE 0.2f

typedef __attribute__((ext_vector_type(2))) float v2f;
typedef __attribute__((ext_vector_type(8))) float v8f;

// Order-preserving float -> uint map for atomicMax-based segment max.
__device__ __forceinline__ unsigned fenc(float x) {
  unsigned u = __float_as_uint(x);
  return (u & 0x80000000u) ? ~u : (u | 0x80000000u);
}
__device__ __forceinline__ float fdec(unsigned v) {
  return (v & 0x80000000u) ? __uint_as_float(v ^ 0x80000000u) : __uint_as_float(~v);
}
__device__ __forceinline__ float lrelu(float x) {
  return x > 0.0f ? x : NEG_SLOPE * x;
}

// ---------------------------------------------------------------------------
// fp32 GEMM strip kernel: C[M, 16*NT] = A[M, KK] * B[KK, 16*NT] (+ bias)
// via V_WMMA_F32_16X16X4_F32.
//  - 128-thread block = 4 waves; wave w computes M-tile (blockIdx.x*4 + w),
//    spanning the FULL N (16*NT columns) with NT accumulators.
//  - Entire B (<= 64 KB) is staged once per block into LDS in a K-PAIR
//    swizzled layout: bs[((k>>1)*N + n)*2 + (k&1)] = B[k][n]. A lane's B
//    fragment (two K rows at the same column) is then one aligned
//    ds_load_b64 straight into the even VGPR pair WMMA wants — no repacking.
// Fragment layouts (ISA 7.12.2, 32-bit):
//  A 16x4: lane(0-15)=row M, v0/v1 = K+0/K+1; lanes 16-31 hold K+2/K+3.
//  B 4x16: lane = col N, v0/v1 = K rows; upper half lanes offset K by 2.
//  C/D 16x16: 8 VGPRs, M = j + 8*(lane>=16), N = lane&15.
// ---------------------------------------------------------------------------
template <int NT, int KK>
__global__ void __launch_bounds__(128) gemm_wmma_strip(
    const float* __restrict__ A, const float* __restrict__ B,
    const float* __restrict__ bias, float* __restrict__ C) {
  constexpr int N = NT * 16;
  __shared__ float bs[KK * N];

  // Cooperative stage of B into LDS, swizzled into K-pairs.
  for (int i = threadIdx.x; i < KK * N; i += 128) {
    const int k = i / N;
    const int n = i % N;
    bs[(((k >> 1) * N) + n) * 2 + (k & 1)] = B[i];
  }
  __syncthreads();

  const int wave = threadIdx.x >> 5;
  const int lane = threadIdx.x & 31;
  const int tm   = blockIdx.x * 4 + wave;
  const int half = lane >> 4;  // 0 or 1
  const int lid  = lane & 15;
  const long mrow = (long)tm * 16 + lid;

  v8f acc[NT];
#pragma unroll
  for (int t = 0; t < NT; ++t) acc[t] = (v8f){};

  const float* __restrict__ Arow = A + mrow * KK;
#pragma unroll 4
  for (int k = 0; k < KK; k += 4) {
    const int kb = k + half * 2;       // even
    const int kp = kb >> 1;            // K-pair index
    v2f a;
    a.x = Arow[kb];
    a.y = Arow[kb + 1];
#pragma unroll
    for (int t = 0; t < NT; ++t) {
      const int ncol = t * 16 + lid;
      const v2f b = *reinterpret_cast<const v2f*>(&bs[(kp * N + ncol) * 2]);
      // (neg_a, A, neg_b, B, c_mod, C, reuse_a, reuse_b)
      acc[t] = __builtin_amdgcn_wmma_f32_16x16x4_f32(false, a, false, b,
                                                     (short)0, acc[t],
                                                     false, false);
    }
  }

  const long row0 = (long)tm * 16 + half * 8;
#pragma unroll
  for (int t = 0; t < NT; ++t) {
    const int ncol = t * 16 + lid;
    const float bv = bias ? bias[ncol] : 0.0f;
#pragma unroll
    for (int j = 0; j < 8; ++j) {
      C[(row0 + j) * N + ncol] = acc[t][j] + bv;
    }
  }
}

// Per (node, head): al_src = <h[n,head,:], a_src[head,:]>, same for a_dst.
__global__ void compute_al_kernel(const float* __restrict__ h,
                                  const float* __restrict__ a_src,
                                  const float* __restrict__ a_dst,
                                  float* __restrict__ al_s,
                                  float* __restrict__ al_d,
                                  int n_nodes, int H, int C) {
  int i = blockIdx.x * blockDim.x + threadIdx.x;
  if (i >= n_nodes * H) return;
  const int head = i % H;
  const float* hp = h + (size_t)i * C;   // [node][head][c] contiguous
  const float* as = a_src + head * C;
  const float* ad = a_dst + head * C;
  float ss = 0.0f, sd = 0.0f;
  for (int c = 0; c < C; ++c) {
    float v = hp[c];
    ss += v * as[c];
    sd += v * ad[c];
  }
  al_s[i] = ss;
  al_d[i] = sd;
}

// Init attention scratch: m -> encoded(-inf), den -> 0, num -> 0.
__global__ void init_attn_kernel(unsigned* __restrict__ m, float* __restrict__ den,
                                 float* __restrict__ num, int n_md, int n_num) {
  int i = blockIdx.x * blockDim.x + threadIdx.x;
  if (i < n_md) {
    m[i] = 0x007FFFFFu;  // fenc(-inf)
    den[i] = 0.0f;
  }
  if (i < n_num) num[i] = 0.0f;
}

// Pass 1 over (edges + self-loops) x heads: segment max of leaky-relu'd logits.
__global__ void edge_max_kernel(const int* __restrict__ src,
                                const int* __restrict__ dst,
                                const float* __restrict__ al_s,
                                const float* __restrict__ al_d,
                                unsigned* __restrict__ mmax,
                                int E, int n_nodes, int H) {
  int i = blockIdx.x * blockDim.x + threadIdx.x;
  int total = (E + n_nodes) * H;
  if (i >= total) return;
  int e = i / H, h = i % H;
  int s, d;
  if (e < E) { s = src[e]; d = dst[e]; } else { s = e - E; d = e - E; }
  float ev = lrelu(al_s[s * H + h] + al_d[d * H + h]);
  atomicMax(&mmax[d * H + h], fenc(ev));
}

// Pass 2: ex = exp(e - m[dst]); den[dst] += ex; num[dst,h,:] += ex * h[src,h,:].
__global__ void edge_accum_kernel(const int* __restrict__ src,
                                  const int* __restrict__ dst,
                                  const float* __restrict__ al_s,
                                  const float* __restrict__ al_d,
                                  const unsigned* __restrict__ mmax,
                                  const float* __restrict__ hmat,
                                  float* __restrict__ den,
                                  float* __restrict__ num,
                                  int E, int n_nodes, int H, int C) {
  int i = blockIdx.x * blockDim.x + threadIdx.x;
  int total = (E + n_nodes) * H;
  if (i >= total) return;
  int e = i / H, h = i % H;
  int s, d;
  if (e < E) { s = src[e]; d = dst[e]; } else { s = e - E; d = e - E; }
  float ev = lrelu(al_s[s * H + h] + al_d[d * H + h]);
  float ex = __expf(ev - fdec(mmax[d * H + h]));
  atomicAdd(&den[d * H + h], ex);
  const float* hs = hmat + (size_t)(s * H + h) * C;
  float* np = num + (size_t)(d * H + h) * C;
  for (int c = 0; c < C; ++c) atomicAdd(&np[c], ex * hs[c]);
}

// out[n, f] = num[n, f] / (den[n, head] + 1e-16) + bias[f]; optional ELU.
__global__ void finalize_kernel(const float* __restrict__ num,
                                const float* __restrict__ den,
                                const float* __restrict__ bias,
                                float* __restrict__ out,
                                int n_nodes, int HC, int H, int C, int do_elu) {
  int i = blockIdx.x * blockDim.x + threadIdx.x;
  if (i >= n_nodes * HC) return;
  int n = i / HC, f = i % HC;
  int head = f / C;
  float v = num[i] / (den[n * H + head] + 1e-16f) + bias[f];
  if (do_elu) v = v > 0.0f ? v : expm1f(v);
  out[i] = v;
}

static inline int cdiv(int a, int b) { return (a + b - 1) / b; }

extern "C" void kernel_launch(void* const* d_in, const int* in_sizes, int n_in,
                              void* d_out, int out_size, void* d_ws, size_t ws_size,
                              hipStream_t stream) {
  // Inputs (setup_inputs order)
  const float* x      = (const float*)d_in[0];
  const int*   ei     = (const int*)d_in[1];  // [2, E] int32 (JAX x64 disabled)
  const float* W1     = (const float*)d_in[2];
  const float* a_src1 = (const float*)d_in[3];
  const float* a_dst1 = (const float*)d_in[4];
  const float* b1     = (const float*)d_in[5];
  const float* W2     = (const float*)d_in[6];
  const float* a_src2 = (const float*)d_in[7];
  const float* a_dst2 = (const float*)d_in[8];
  const float* b2     = (const float*)d_in[9];
  const float* W_out  = (const float*)d_in[10];
  const float* b_out  = (const float*)d_in[11];
  float* out = (float*)d_out;

  const int* src = ei;            // row 0
  const int* dst = ei + N_EDGES;  // row 1

  const int HC1 = HEADS * HID;    // 128
  const int HC2 = HID;            // 32

  // Workspace layout (floats). h1/h2 are activated in place after aggregation.
  float* ws = (float*)d_ws;
  size_t off = 0;
  float*    h1   = ws + off; off += (size_t)N_NODES * HC1;   // 5.12M
  float*    num1 = ws + off; off += (size_t)N_NODES * HC1;   // reused as num2
  float*    h2   = ws + off; off += (size_t)N_NODES * HC2;
  float*    als1 = ws + off; off += (size_t)N_NODES * HEADS;
  float*    ald1 = ws + off; off += (size_t)N_NODES * HEADS;
  unsigned* m1   = (unsigned*)(ws + off); off += (size_t)N_NODES * HEADS;
  float*    den1 = ws + off; off += (size_t)N_NODES * HEADS;
  float*    als2 = ws + off; off += (size_t)N_NODES;
  float*    ald2 = ws + off; off += (size_t)N_NODES;
  unsigned* m2   = (unsigned*)(ws + off); off += (size_t)N_NODES;
  float*    den2 = ws + off; off += (size_t)N_NODES;
  float*    num2 = num1;

  const int TB = 256;
  const int EH1 = (N_EDGES + N_NODES) * HEADS;
  const int EH2 = (N_EDGES + N_NODES) * 1;
  const int MBLK = (N_NODES / 16) / 4;   // 625 blocks, 4 M-tiles (waves) each

  // ---- Layer 1: h1 = x @ W1  (40000x128 @ 128x128) ----
  gemm_wmma_strip<8, IN_CH><<<MBLK, 128, 0, stream>>>(x, W1, nullptr, h1);
  compute_al_kernel<<<cdiv(N_NODES * HEADS, TB), TB, 0, stream>>>(
      h1, a_src1, a_dst1, als1, ald1, N_NODES, HEADS, HID);
  init_attn_kernel<<<cdiv(N_NODES * HC1, TB), TB, 0, stream>>>(
      m1, den1, num1, N_NODES * HEADS, N_NODES * HC1);
  edge_max_kernel<<<cdiv(EH1, TB), TB, 0, stream>>>(
      src, dst, als1, ald1, m1, N_EDGES, N_NODES, HEADS);
  edge_accum_kernel<<<cdiv(EH1, TB), TB, 0, stream>>>(
      src, dst, als1, ald1, m1, h1, den1, num1, N_EDGES, N_NODES, HEADS, HID);
  // ELU(aggregate + b1) written in place over h1
  finalize_kernel<<<cdiv(N_NODES * HC1, TB), TB, 0, stream>>>(
      num1, den1, b1, h1, N_NODES, HC1, HEADS, HID, 1);

  // ---- Layer 2: h2 = h1 @ W2  (40000x128 @ 128x32) ----
  gemm_wmma_strip<2, 128><<<MBLK, 128, 0, stream>>>(h1, W2, nullptr, h2);
  compute_al_kernel<<<cdiv(N_NODES, TB), TB, 0, stream>>>(
      h2, a_src2, a_dst2, als2, ald2, N_NODES, 1, HID);
  init_attn_kernel<<<cdiv(N_NODES * HC2, TB), TB, 0, stream>>>(
      m2, den2, num2, N_NODES, N_NODES * HC2);
  edge_max_kernel<<<cdiv(EH2, TB), TB, 0, stream>>>(
      src, dst, als2, ald2, m2, N_EDGES, N_NODES, 1);
  edge_accum_kernel<<<cdiv(EH2, TB), TB, 0, stream>>>(
      src, dst, als2, ald2, m2, h2, den2, num2, N_EDGES, N_NODES, 1, HID);
  finalize_kernel<<<cdiv(N_NODES * HC2, TB), TB, 0, stream>>>(
      num2, den2, b2, h2, N_NODES, HC2, 1, HID, 1);

  // ---- Output projection: out = h2 @ W_out + b_out  (40000x32 @ 32x112) ----
  gemm_wmma_strip<7, HID><<<MBLK, 128, 0, stream>>>(h2, W_out, b_out, out);
}